// LSTM_71167608095114
// MI455X (gfx1250) — compile-verified
//
#include <hip/hip_runtime.h>
#include <hip/hip_bf16.h>

// ---------------------------------------------------------------------------
// 2-layer LSTM (T=1024, B=64, I=H=512), fp32, for MI455X (gfx1250).
//
// Design:
//  * Persistent kernel: 32 workgroups x 256 threads (8 wave32 waves).
//  * Gate matrix is [64 x 2048] per step; WG `wg` owns hidden columns
//    [wg*16, wg*16+16) => gate column tiles {wg, wg+32, wg+64, wg+96}.
//  * Per-WG weight slab (Wih rows + Whh rows for its 4 gate tiles,
//    K = I+H = 1024, N = 16) = 256 KB, resident in LDS for the whole layer,
//    staged with GLOBAL_LOAD_ASYNC_TO_LDS_B128 (ASYNCcnt / s_wait_asynccnt).
//  * Waves 0-3: row tiles m=0..3 for gates i,f.  Waves 4-7: same rows for
//    gates g,o.  Each wave: two v8f f32 accumulators; V_WMMA_F32_16X16X4_F32
//    with K stepped by 4 over 1024.
//  * c-state lives in VGPRs of waves 0-3 for the entire scan.
//  * Grid-wide sync: monotonic atomic counter (2 barriers per step) so
//    layer 1 can read+overwrite the y sequence in d_out in place.
// ---------------------------------------------------------------------------

typedef float v2f __attribute__((ext_vector_type(2)));
typedef float v8f __attribute__((ext_vector_type(8)));
typedef int   v4i_t __attribute__((vector_size(4 * sizeof(int))));
typedef v4i_t __attribute__((address_space(1)))* gptr_v4i;   // global (AS1)
typedef v4i_t __attribute__((address_space(3)))* lptr_v4i;   // LDS (AS3)

#define T_STEPS 1024
#define BATCH   64
#define ISZ     512
#define HSZ     512
#define GSZ     (4 * HSZ)          // 2048
#define NLAYER  2
#define NWG     32
#define KBLK_X  128                // k-steps of 4 over input half
#define KBLK_H  128                // k-steps of 4 over hidden half
#define KROW    64                 // floats per k-step row in LDS slab (16 n * 4 kk)
#define LDS_W_FLOATS (4 * (KBLK_X + KBLK_H) * KROW)   // 65536 floats = 256 KB
#define LDS_X_FLOATS (4 * 2 * 256)                    // 2048 floats  =   8 KB

// ---- CDNA5 async global->LDS copy (ASYNCcnt-tracked, no VGPR data) --------
__device__ __forceinline__ void async_g2l_b128(void* lds_dst, const void* gsrc) {
#if __has_builtin(__builtin_amdgcn_global_load_async_to_lds_b128)
  __builtin_amdgcn_global_load_async_to_lds_b128(
      (gptr_v4i)(uintptr_t)gsrc,
      (lptr_v4i)(unsigned)(uintptr_t)lds_dst,
      0, 0);
#else
  unsigned dst = (unsigned)(uintptr_t)lds_dst;   // LDS byte offset (addr[31:0])
  asm volatile("global_load_async_to_lds_b128 %0, %1, off"
               :: "v"(dst), "v"(gsrc) : "memory");
#endif
}

__device__ __forceinline__ void wait_async0() {
#if __has_builtin(__builtin_amdgcn_s_wait_asynccnt)
  __builtin_amdgcn_s_wait_asynccnt(0);
#else
  asm volatile("s_wait_asynccnt 0" ::: "memory");
#endif
}

__device__ __forceinline__ void grid_sync(unsigned* cnt, unsigned goal) {
  __threadfence();
  __syncthreads();
  if (threadIdx.x == 0) {
    __hip_atomic_fetch_add(cnt, 1u, __ATOMIC_ACQ_REL, __HIP_MEMORY_SCOPE_AGENT);
    while (__hip_atomic_load(cnt, __ATOMIC_ACQUIRE, __HIP_MEMORY_SCOPE_AGENT) < goal)
      __builtin_amdgcn_s_sleep(2);
  }
  __syncthreads();
}

__device__ __forceinline__ float sigmoidf_(float x) {
  return 1.0f / (1.0f + __expf(-x));
}

__global__ void __launch_bounds__(256, 1)
lstm_scan_kernel(const float* __restrict__ x_in,
                 const float* __restrict__ h0,
                 const float* __restrict__ c0,
                 const float* __restrict__ Wih,
                 const float* __restrict__ Whh,
                 const float* __restrict__ bih,
                 const float* __restrict__ bhh,
                 float* __restrict__ out,
                 float* __restrict__ h_ws,
                 unsigned* __restrict__ counter)
{
  extern __shared__ float lds[];
  float* ldsW = lds;                   // [gate][kb][n][kk]
  float* ldsX = lds + LDS_W_FLOATS;    // [m][{g,o}][8 vgpr x 32 lane]

  const int tid  = threadIdx.x;
  const int lane = tid & 31;
  const int wave = tid >> 5;           // 0..7
  const int m    = wave & 3;           // row tile (16 rows each)
  const int gp   = wave >> 2;          // 0 -> gates i,f ; 1 -> gates g,o
  const int wg   = blockIdx.x;         // hidden-column tile 0..31

  const int nloc  = lane & 15;                 // B/C/D column within tile
  const int koff  = (lane >> 4) << 1;          // A/B k sub-offset: 0 or 2
  const int arow  = m * 16 + nloc;             // A-matrix row (M = lane%16)
  const int rbase = m * 16 + ((lane >> 4) << 3); // C/D row base (lanes>=16 -> +8)
  const int col   = wg * 16 + nloc;            // hidden column

  float* yout = out;                                     // [T,B,H]
  float* hT   = out + (size_t)T_STEPS * BATCH * HSZ;     // [L,B,H]
  float* cT   = hT + (size_t)NLAYER * BATCH * HSZ;       // [L,B,H]

  unsigned bar = 0;

  for (int l = 0; l < NLAYER; ++l) {
    // ---- Async-stage this WG's weight slab into LDS: [gate][kb][n][kk] ----
    // One b128 transfer moves 4 consecutive k-values of one weight row into
    // their contiguous 16-byte LDS slot. Consecutive lanes take consecutive
    // kb of the same row -> 512 B coalesced global reads per wave issue.
    const float* WihL = Wih + (size_t)l * GSZ * ISZ;
    const float* WhhL = Whh + (size_t)l * GSZ * HSZ;
    for (int it = 0; it < 64; ++it) {
      int id = it * 256 + tid;          // 16384 transfers total
      int kb = id & 255;
      int n  = (id >> 8) & 15;
      int g  = id >> 12;
      int k  = kb * 4;
      int grow = g * HSZ + wg * 16 + n;
      const float* src = (k < ISZ) ? (WihL + (size_t)grow * ISZ + k)
                                   : (WhhL + (size_t)grow * HSZ + (k - ISZ));
      async_g2l_b128(ldsW + (size_t)((g * 256 + kb) * 16 + n) * 4, src);
    }

    // ---- Combined biases for this wave's two gates at column nloc ---------
    const int g0 = gp * 2, g1 = gp * 2 + 1;
    const float bias0 = bih[l * GSZ + g0 * HSZ + wg * 16 + nloc]
                      + bhh[l * GSZ + g0 * HSZ + wg * 16 + nloc];
    const float bias1 = bih[l * GSZ + g1 * HSZ + wg * 16 + nloc]
                      + bhh[l * GSZ + g1 * HSZ + wg * 16 + nloc];

    // ---- Init h (this WG's columns) and register-resident c ---------------
    for (int idx = tid; idx < BATCH * 16; idx += 256) {
      int b = idx >> 4, cc = idx & 15;
      h_ws[b * HSZ + wg * 16 + cc] =
          h0[(size_t)l * BATCH * HSZ + b * HSZ + wg * 16 + cc];
    }
    v8f creg = {0.f, 0.f, 0.f, 0.f, 0.f, 0.f, 0.f, 0.f};
    if (gp == 0) {
      for (int r = 0; r < 8; ++r)
        creg[r] = c0[(size_t)l * BATCH * HSZ + (size_t)(rbase + r) * HSZ + col];
    }

    wait_async0();                      // weight slab resident in LDS
    grid_sync(counter, ++bar * NWG);

    const float* xsrc = (l == 0) ? x_in : yout;   // layer 1: in-place on d_out
    const float* bp0 = ldsW + g0 * ((KBLK_X + KBLK_H) * KROW) + nloc * 4 + koff;
    const float* bp1 = ldsW + g1 * ((KBLK_X + KBLK_H) * KROW) + nloc * 4 + koff;

    for (int t = 0; t < T_STEPS; ++t) {
      v8f acc0 = {bias0, bias0, bias0, bias0, bias0, bias0, bias0, bias0};
      v8f acc1 = {bias1, bias1, bias1, bias1, bias1, bias1, bias1, bias1};
      const float* xrow = xsrc + ((size_t)t * BATCH + arow) * ISZ + koff;
      const float* hrow = h_ws + (size_t)arow * HSZ + koff;

      if (t + 1 < T_STEPS)   // warm L2/WGP$ for the next step's A stream
        __builtin_prefetch(xsrc + ((size_t)(t + 1) * BATCH + arow) * ISZ, 0, 0);

      #pragma unroll 4
      for (int kb = 0; kb < KBLK_X; ++kb) {           // input half of K
        v2f a  = *(const v2f*)(xrow + kb * 4);
        v2f b0 = *(const v2f*)(bp0 + kb * KROW);
        v2f b1 = *(const v2f*)(bp1 + kb * KROW);
        acc0 = __builtin_amdgcn_wmma_f32_16x16x4_f32(false, a, false, b0,
                                                     (short)0, acc0, false, false);
        acc1 = __builtin_amdgcn_wmma_f32_16x16x4_f32(false, a, false, b1,
                                                     (short)0, acc1, false, false);
      }
      #pragma unroll 4
      for (int kb = 0; kb < KBLK_H; ++kb) {           // hidden half of K
        v2f a  = *(const v2f*)(hrow + kb * 4);
        v2f b0 = *(const v2f*)(bp0 + (KBLK_X + kb) * KROW);
        v2f b1 = *(const v2f*)(bp1 + (KBLK_X + kb) * KROW);
        acc0 = __builtin_amdgcn_wmma_f32_16x16x4_f32(false, a, false, b0,
                                                     (short)0, acc0, false, false);
        acc1 = __builtin_amdgcn_wmma_f32_16x16x4_f32(false, a, false, b1,
                                                     (short)0, acc1, false, false);
      }

      grid_sync(counter, ++bar * NWG);  // all reads of x[t]/h done everywhere

      // Exchange g,o accumulators (waves 4-7) to waves 0-3 via LDS.
      if (gp == 1) {
        float* xg = ldsX + m * 512;
        for (int r = 0; r < 8; ++r) {
          xg[r * 32 + lane]       = acc0[r];   // g gate pre-activation
          xg[256 + r * 32 + lane] = acc1[r];   // o gate pre-activation
        }
      }
      __syncthreads();
      if (gp == 0) {
        const float* xg = ldsX + m * 512;
        for (int r = 0; r < 8; ++r) {
          float iv = sigmoidf_(acc0[r]);
          float fv = sigmoidf_(acc1[r]);
          float gv = tanhf(xg[r * 32 + lane]);
          float ov = sigmoidf_(xg[256 + r * 32 + lane]);
          float cv = fv * creg[r] + iv * gv;
          creg[r] = cv;
          float hv = ov * tanhf(cv);
          int row = rbase + r;
          h_ws[(size_t)row * HSZ + col] = hv;
          yout[((size_t)t * BATCH + row) * HSZ + col] = hv;
          if (t == T_STEPS - 1)
            hT[(size_t)l * BATCH * HSZ + (size_t)row * HSZ + col] = hv;
        }
      }
      grid_sync(counter, ++bar * NWG);  // h_new visible before next step
    }

    if (gp == 0) {
      for (int r = 0; r < 8; ++r)
        cT[(size_t)l * BATCH * HSZ + (size_t)(rbase + r) * HSZ + col] = creg[r];
    }
    grid_sync(counter, ++bar * NWG);    // layer boundary
  }
}

extern "C" void kernel_launch(void* const* d_in, const int* in_sizes, int n_in,
                              void* d_out, int out_size, void* d_ws, size_t ws_size,
                              hipStream_t stream) {
  (void)in_sizes; (void)n_in; (void)out_size; (void)ws_size;
  const float* x   = (const float*)d_in[0];
  const float* h0  = (const float*)d_in[1];
  const float* c0  = (const float*)d_in[2];
  const float* Wih = (const float*)d_in[3];
  const float* Whh = (const float*)d_in[4];
  const float* bih = (const float*)d_in[5];
  const float* bhh = (const float*)d_in[6];
  float* out = (float*)d_out;

  float*    h_ws    = (float*)d_ws;                                  // B*H f32
  unsigned* counter = (unsigned*)((char*)d_ws + (size_t)BATCH * HSZ * sizeof(float));

  (void)hipMemsetAsync(counter, 0, sizeof(unsigned), stream);  // graph-capturable

  size_t shmem = (size_t)(LDS_W_FLOATS + LDS_X_FLOATS) * sizeof(float); // 264 KB
  lstm_scan_kernel<<<dim3(NWG), dim3(256), shmem, stream>>>(
      x, h0, c0, Wih, Whh, bih, bhh, out, h_ws, counter);
}